// GGNN_9826885173933
// MI455X (gfx1250) — compile-verified
//
#include <hip/hip_runtime.h>

#define N_NODES 50000
#define N_EDGES 800000
#define HDIM    64
#define H3      192
#define KTYPES  4
#define NSTEPS  4

typedef float v2f __attribute__((ext_vector_type(2)));
typedef float v8f __attribute__((ext_vector_type(8)));

// --------------------------------------------------------------------------
// transformed[k][n][h] = sum_d feat[n][d] * W[k][h][d] + b[k][h]
// One block = 16 nodes, 4 waves; wave w handles etype w, all 64 output cols.
// WMMA f32 16x16x4: A frag (lane&15 = M, lane>>4 selects K pair), B frag
// (lane&15 = N col, lane>>4 selects K pair) -> both are float2 loads.
// --------------------------------------------------------------------------
__global__ __launch_bounds__(128) void k_transform(
    const float* __restrict__ feat, const float* __restrict__ W,
    const float* __restrict__ bias, float* __restrict__ out) {
  const int tile  = blockIdx.x;           // node tile (16 rows)
  const int k     = threadIdx.x >> 5;     // etype = wave id
  const int lane  = threadIdx.x & 31;
  const int m0    = tile * 16;
  const int mrow  = lane & 15;
  const int khalf = lane >> 4;            // 0 or 1
  const int nc    = lane & 15;

  const v8f vz = {0.f,0.f,0.f,0.f,0.f,0.f,0.f,0.f};
  v8f acc[4];
#pragma unroll
  for (int t = 0; t < 4; ++t) acc[t] = vz;

  const float* Arow = feat + (size_t)(m0 + mrow) * HDIM;
  const float* Wk   = W + (size_t)k * HDIM * HDIM;

#pragma unroll
  for (int kk = 0; kk < HDIM; kk += 4) {
    const int d0 = kk + khalf * 2;
    v2f afrag = *(const v2f*)(Arow + d0);
#pragma unroll
    for (int t = 0; t < 4; ++t) {
      const int col = t * 16 + nc;
      v2f bfrag = *(const v2f*)(Wk + (size_t)col * HDIM + d0);  // B[d][col]=W[k][col][d]
      acc[t] = __builtin_amdgcn_wmma_f32_16x16x4_f32(
          false, afrag, false, bfrag, (short)0, acc[t], false, false);
    }
  }

  // D layout: VGPR i -> row i (lanes 0-15) / row i+8 (lanes 16-31), col = lane&15
  float* outk = out + ((size_t)k * N_NODES + m0) * HDIM;
  const int mofs = khalf * 8;
#pragma unroll
  for (int t = 0; t < 4; ++t) {
    const int col = t * 16 + nc;
    const float bv = bias[k * HDIM + col];
#pragma unroll
    for (int i = 0; i < 8; ++i)
      outk[(size_t)(mofs + i) * HDIM + col] = acc[t][i] + bv;
  }
}

// --------------------------------------------------------------------------
__global__ void k_zero(float4* __restrict__ a, int n4) {
  int i = blockIdx.x * blockDim.x + threadIdx.x;
  if (i < n4) a[i] = make_float4(0.f, 0.f, 0.f, 0.f);
}

// --------------------------------------------------------------------------
// a[dst[e]][:] += transformed[etypes[e]][src[e]][:]   (16 threads per edge)
// --------------------------------------------------------------------------
__global__ __launch_bounds__(256) void k_scatter(
    const float* __restrict__ tr, const int* __restrict__ src,
    const int* __restrict__ dst, const int* __restrict__ et,
    float* __restrict__ a) {
  const int t = blockIdx.x * blockDim.x + threadIdx.x;
  const int e = t >> 4;
  if (e >= N_EDGES) return;
  const int c = (t & 15) * 4;
  const int s = src[e], d = dst[e], k = et[e];
  const float4 m = *(const float4*)(tr + ((size_t)k * N_NODES + s) * HDIM + c);
  float* ap = a + (size_t)d * HDIM + c;
  atomicAdd(ap + 0, m.x);
  atomicAdd(ap + 1, m.y);
  atomicAdd(ap + 2, m.z);
  atomicAdd(ap + 3, m.w);
}

// --------------------------------------------------------------------------
// GRU cell for a 16-node tile.
// gi = agg @ w_ih^T + b_ih  (waves 0,1)   gh = feat @ w_hh^T + b_hh (waves 2,3)
// Each wave: 6 col-tiles of 16 (96 cols), 16-deep K loop, 96 WMMAs.
// Results staged in LDS, then 128 threads do the gate math.
// --------------------------------------------------------------------------
__global__ __launch_bounds__(128) void k_gru(
    const float* __restrict__ agg, const float* __restrict__ feat,
    const float* __restrict__ w_ih, const float* __restrict__ w_hh,
    const float* __restrict__ b_ih, const float* __restrict__ b_hh,
    float* __restrict__ out) {
  __shared__ float lds[2][16][H3];   // [0]=gi, [1]=gh : 24 KB

  const int tile    = blockIdx.x;
  const int wave    = threadIdx.x >> 5;
  const int lane    = threadIdx.x & 31;
  const int m0      = tile * 16;
  const int sel     = wave >> 1;             // 0 -> gi, 1 -> gh
  const int colbase = (wave & 1) * 96;
  const int mrow    = lane & 15;
  const int khalf   = lane >> 4;
  const int nc      = lane & 15;

  const float* A  = sel ? feat : agg;
  const float* B  = sel ? w_hh : w_ih;
  const float* bb = sel ? b_hh : b_ih;

  const v8f vz = {0.f,0.f,0.f,0.f,0.f,0.f,0.f,0.f};
  v8f acc[6];
#pragma unroll
  for (int t = 0; t < 6; ++t) acc[t] = vz;

  const float* Arow = A + (size_t)(m0 + mrow) * HDIM;
#pragma unroll
  for (int kk = 0; kk < HDIM; kk += 4) {
    const int d0 = kk + khalf * 2;
    v2f afrag = *(const v2f*)(Arow + d0);
#pragma unroll
    for (int t = 0; t < 6; ++t) {
      const int col = colbase + t * 16 + nc;
      v2f bfrag = *(const v2f*)(B + (size_t)col * HDIM + d0);  // B[d][col]=w[col][d]
      acc[t] = __builtin_amdgcn_wmma_f32_16x16x4_f32(
          false, afrag, false, bfrag, (short)0, acc[t], false, false);
    }
  }

  const int mofs = khalf * 8;
#pragma unroll
  for (int t = 0; t < 6; ++t) {
    const int col = colbase + t * 16 + nc;
    const float bv = bb[col];
#pragma unroll
    for (int i = 0; i < 8; ++i)
      lds[sel][mofs + i][col] = acc[t][i] + bv;
  }
  __syncthreads();

  // gate math: r,z,n over 16 nodes x 64 features
  for (int idx = threadIdx.x; idx < 16 * HDIM; idx += 128) {
    const int n = idx >> 6;
    const int j = idx & 63;
    const float xr  = lds[0][n][j]        + lds[1][n][j];
    const float xz  = lds[0][n][j + 64]   + lds[1][n][j + 64];
    const float i_n = lds[0][n][j + 128];
    const float h_n = lds[1][n][j + 128];
    const float r = 1.f / (1.f + __expf(-xr));
    const float z = 1.f / (1.f + __expf(-xz));
    const float tv = i_n + r * h_n;
    const float e2 = __expf(2.f * tv);
    const float nn = (e2 - 1.f) / (e2 + 1.f);           // tanh
    const float f  = feat[(size_t)(m0 + n) * HDIM + j];
    out[(size_t)(m0 + n) * HDIM + j] = (1.f - z) * nn + z * f;
  }
}

// --------------------------------------------------------------------------
extern "C" void kernel_launch(void* const* d_in, const int* in_sizes, int n_in,
                              void* d_out, int out_size, void* d_ws, size_t ws_size,
                              hipStream_t stream) {
  const float* h    = (const float*)d_in[0];
  const float* W    = (const float*)d_in[1];
  const float* b    = (const float*)d_in[2];
  const float* w_ih = (const float*)d_in[3];
  const float* w_hh = (const float*)d_in[4];
  const float* b_ih = (const float*)d_in[5];
  const float* b_hh = (const float*)d_in[6];
  const int*   src  = (const int*)d_in[7];
  const int*   dst  = (const int*)d_in[8];
  const int*   et   = (const int*)d_in[9];
  float* out = (float*)d_out;

  const size_t NH = (size_t)N_NODES * HDIM;
  float* tr  = (float*)d_ws;                 // [K][N][H] : 51.2 MB
  float* agg = tr + (size_t)KTYPES * NH;     // [N][H]    : 12.8 MB
  float* fA  = agg + NH;                     // [N][H]
  float* fB  = fA + NH;                      // [N][H]    total ~89.6 MB

  const dim3 tiles(N_NODES / 16);            // 3125, exact
  const dim3 tb128(128);
  const dim3 zgrid((N_NODES * HDIM / 4 + 255) / 256);
  const dim3 sgrid((size_t)N_EDGES * 16 / 256);
  const dim3 tb256(256);

  const float* cur = h;
  for (int s = 0; s < NSTEPS; ++s) {
    float* nxt = (s == NSTEPS - 1) ? out : (((s & 1) == 0) ? fA : fB);

    k_transform<<<tiles, tb128, 0, stream>>>(cur, W, b, tr);
    k_zero<<<zgrid, tb256, 0, stream>>>((float4*)agg, N_NODES * HDIM / 4);
    k_scatter<<<sgrid, tb256, 0, stream>>>(tr, src, dst, et, agg);
    k_gru<<<tiles, tb128, 0, stream>>>(agg, cur, w_ih, w_hh, b_ih, b_hh, nxt);

    cur = nxt;
  }
}